// ModelNew_17514876633353
// MI455X (gfx1250) — compile-verified
//
#include <hip/hip_runtime.h>
#include <hip/hip_bf16.h>

typedef __attribute__((ext_vector_type(2))) float v2f;
typedef __attribute__((ext_vector_type(8))) float v8f;

#define B_  64
#define N_  2048
#define D_  512
#define K_  64
#define KG_ 80
#define ROWS_ (B_ * N_)          // 131072

// ---------------------------------------------------------------------------
// Kernel 1: a_raw = x @ clusters  (131072x512 @ 512x80), f32 WMMA 16x16x4.
// Also emits per-block column sum / sum-of-squares partials for BatchNorm.
// Block = 256 threads = 8 waves; each wave owns 16 rows x 80 cols.
// ---------------------------------------------------------------------------
__global__ __launch_bounds__(256) void k1_gemm1(
    const float* __restrict__ x, const float* __restrict__ clusters,
    float* __restrict__ a_raw, float* __restrict__ p_sum,
    float* __restrict__ p_ssq) {
  const int lane = threadIdx.x & 31;
  const int wave = threadIdx.x >> 5;
  const int l15  = lane & 15;
  const int hs   = lane >> 4;                  // 0: K={0,1}  1: K={2,3}
  const int r0   = blockIdx.x * 128 + wave * 16;

  v8f acc[5];
#pragma unroll
  for (int t = 0; t < 5; ++t) acc[t] = (v8f){0.f,0.f,0.f,0.f,0.f,0.f,0.f,0.f};

  // A fragment source: row = r0 + l15, columns k0 + 2*hs + {0,1}
  const float* xrow = x + (size_t)(r0 + l15) * D_ + hs * 2;

  for (int k0 = 0; k0 < D_; k0 += 4) {
    v2f a = *(const v2f*)(xrow + k0);          // 8B aligned
#pragma unroll
    for (int t = 0; t < 5; ++t) {
      const float* bp = clusters + (size_t)(k0 + hs * 2) * KG_ + t * 16 + l15;
      v2f b;
      b.x = bp[0];
      b.y = bp[KG_];
      acc[t] = __builtin_amdgcn_wmma_f32_16x16x4_f32(
          false, a, false, b, (short)0, acc[t], false, false);
    }
  }

  // Store a_raw + per-wave column partial sums (deterministic order)
  __shared__ float s_sum[8][KG_];
  __shared__ float s_ssq[8][KG_];
#pragma unroll
  for (int t = 0; t < 5; ++t) {
    float s = 0.f, q = 0.f;
#pragma unroll
    for (int r = 0; r < 8; ++r) {
      float v = acc[t][r];
      int row = r0 + r + hs * 8;               // C layout: M = r (+8 upper half)
      a_raw[(size_t)row * KG_ + t * 16 + l15] = v;
      s += v;
      q += v * v;
    }
    s += __shfl_xor(s, 16, 32);                // combine the two M halves
    q += __shfl_xor(q, 16, 32);
    if (hs == 0) { s_sum[wave][t * 16 + l15] = s; s_ssq[wave][t * 16 + l15] = q; }
  }
  __syncthreads();
  int c = threadIdx.x;
  if (c < KG_) {
    float s = 0.f, q = 0.f;
#pragma unroll
    for (int w = 0; w < 8; ++w) { s += s_sum[w][c]; q += s_ssq[w][c]; }
    p_sum[(size_t)blockIdx.x * KG_ + c] = s;
    p_ssq[(size_t)blockIdx.x * KG_ + c] = q;
  }
}

// ---------------------------------------------------------------------------
// Kernel 2: finalize BN stats -> fused scale/shift  (80 columns, 1 block)
// ---------------------------------------------------------------------------
__global__ void k2_bnstats(const float* __restrict__ p_sum,
                           const float* __restrict__ p_ssq,
                           const float* __restrict__ bnw,
                           const float* __restrict__ bnb,
                           float* __restrict__ scale_shift, int nblocks) {
  int c = threadIdx.x;
  if (c >= KG_) return;
  float s = 0.f, q = 0.f;
  for (int i = 0; i < nblocks; ++i) {
    s += p_sum[(size_t)i * KG_ + c];
    q += p_ssq[(size_t)i * KG_ + c];
  }
  const float inv_n = 1.0f / (float)ROWS_;
  float mean = s * inv_n;
  float var  = q * inv_n - mean * mean;        // biased variance
  float sc   = bnw[c] * rsqrtf(var + 1e-5f);
  scale_shift[c]        = sc;
  scale_shift[KG_ + c]  = bnb[c] - mean * sc;
}

// ---------------------------------------------------------------------------
// Kernel 3: per-row affine + softmax(80), keep first 64 -> assignment
// One thread per row.
// ---------------------------------------------------------------------------
__global__ __launch_bounds__(256) void k3_softmax(
    const float* __restrict__ a_raw, const float* __restrict__ ss,
    float* __restrict__ assign) {
  int r = blockIdx.x * 256 + threadIdx.x;
  const float* ar = a_raw + (size_t)r * KG_;
  float v[KG_];
  float mx = -3.402823466e+38f;
#pragma unroll
  for (int i = 0; i < KG_; ++i) {
    v[i] = ar[i] * ss[i] + ss[KG_ + i];
    mx = fmaxf(mx, v[i]);
  }
  float sum = 0.f;
#pragma unroll
  for (int i = 0; i < KG_; ++i) {
    v[i] = __expf(v[i] - mx);
    sum += v[i];
  }
  float inv = 1.0f / sum;
  float* out = assign + (size_t)r * K_;
#pragma unroll
  for (int i = 0; i < K_; ++i) out[i] = v[i] * inv;
}

// ---------------------------------------------------------------------------
// Kernel 3b: a_sum[b][k] = sum_n assignment[b][n][k]   (one block per b)
// ---------------------------------------------------------------------------
__global__ __launch_bounds__(256) void k3b_asum(const float* __restrict__ assign,
                                                float* __restrict__ a_sum) {
  int b = blockIdx.x, t = threadIdx.x;
  int k = t & 63, chunk = t >> 6;              // 4 chunks of 512 rows
  const float* p = assign + (size_t)b * N_ * K_ + (size_t)chunk * 512 * K_ + k;
  float s = 0.f;
  for (int n = 0; n < 512; ++n) s += p[(size_t)n * K_];
  __shared__ float sp[256];
  sp[t] = s;
  __syncthreads();
  if (t < 64)
    a_sum[b * K_ + t] = sp[t] + sp[t + 64] + sp[t + 128] + sp[t + 192];
}

// ---------------------------------------------------------------------------
// Kernel 4: vlad[b] = x[b]^T @ assignment[b] - a_sum[b]*clusters2
// f32 WMMA 16x16x4 with N as the contraction dim. Block = 8 waves covering
// 32 d-rows x 64 k-cols; grid = 64 b x 16 d-groups. Writes pre-norm to d_out.
// ---------------------------------------------------------------------------
__global__ __launch_bounds__(256) void k4_vlad(
    const float* __restrict__ x, const float* __restrict__ assign,
    const float* __restrict__ a_sum, const float* __restrict__ clusters2,
    float* __restrict__ out) {
  int b    = blockIdx.x >> 4;
  int dgrp = blockIdx.x & 15;
  int wave = threadIdx.x >> 5, lane = threadIdx.x & 31;
  int l15 = lane & 15, hs = lane >> 4;
  int d0 = (dgrp * 2 + (wave >> 2)) * 16;
  int k0 = (wave & 3) * 16;

  const float* xb = x + (size_t)b * N_ * D_;
  const float* ab = assign + (size_t)b * N_ * K_;

  v8f acc = (v8f){0.f,0.f,0.f,0.f,0.f,0.f,0.f,0.f};
  for (int n0 = 0; n0 < N_; n0 += 4) {
    int n = n0 + hs * 2;
    v2f a, bb;
    a.x  = xb[(size_t)n * D_ + d0 + l15];       // A[m][k] = x[n0+k][d0+m]
    a.y  = xb[(size_t)(n + 1) * D_ + d0 + l15];
    bb.x = ab[(size_t)n * K_ + k0 + l15];       // B[k][n] = assign[n0+k][k0+n]
    bb.y = ab[(size_t)(n + 1) * K_ + k0 + l15];
    __builtin_prefetch(xb + (size_t)(n0 + 8) * D_ + d0 + l15, 0, 1);
    acc = __builtin_amdgcn_wmma_f32_16x16x4_f32(
        false, a, false, bb, (short)0, acc, false, false);
  }
#pragma unroll
  for (int r = 0; r < 8; ++r) {
    int d = d0 + r + hs * 8;
    int k = k0 + l15;
    float v = acc[r] - a_sum[b * K_ + k] * clusters2[(size_t)d * K_ + k];
    out[(size_t)b * (D_ * K_) + (size_t)d * K_ + k] = v;
  }
}

// ---------------------------------------------------------------------------
// Kernel 5: intra-normalize over D, then global L2 normalize, in place.
// One block per b. Global sumsq derived from per-k sumsq (no extra pass).
// ---------------------------------------------------------------------------
__global__ __launch_bounds__(256) void k5_norm(float* __restrict__ out) {
  int b = blockIdx.x, t = threadIdx.x;
  int k = t & 63, dc = t >> 6;                 // 4 d-chunks of 128
  float* vb = out + (size_t)b * (D_ * K_);
  float s = 0.f;
  for (int d = dc * 128; d < dc * 128 + 128; ++d) {
    float v = vb[(size_t)d * K_ + k];
    s += v * v;
  }
  __shared__ float sp[256];
  __shared__ float sinv[64];
  __shared__ float gpart[64];
  __shared__ float ginv_s;
  sp[t] = s;
  __syncthreads();
  if (t < 64) {
    float sq = sp[t] + sp[t + 64] + sp[t + 128] + sp[t + 192];
    float inv = 1.0f / fmaxf(sqrtf(sq), 1e-12f);
    sinv[t]  = inv;
    gpart[t] = sq * inv * inv;                 // post-intra-norm sumsq
  }
  __syncthreads();
  if (t == 0) {
    float g = 0.f;
#pragma unroll
    for (int i = 0; i < 64; ++i) g += gpart[i];
    ginv_s = 1.0f / fmaxf(sqrtf(g), 1e-12f);
  }
  __syncthreads();
  float scale = sinv[k] * ginv_s;
  for (int d = dc * 128; d < dc * 128 + 128; ++d)
    vb[(size_t)d * K_ + k] *= scale;
}

// ---------------------------------------------------------------------------
extern "C" void kernel_launch(void* const* d_in, const int* in_sizes, int n_in,
                              void* d_out, int out_size, void* d_ws, size_t ws_size,
                              hipStream_t stream) {
  const float* x        = (const float*)d_in[0];   // (64,2048,512)
  const float* clusters = (const float*)d_in[1];   // (512,80)
  const float* clusters2= (const float*)d_in[2];   // (1,512,64)
  const float* bnw      = (const float*)d_in[3];   // (80,)
  const float* bnb      = (const float*)d_in[4];   // (80,)
  float* out = (float*)d_out;                      // (64, 512*64)

  const int nblocks1 = ROWS_ / 128;                // 1024

  float* ws = (float*)d_ws;
  size_t off = 0;
  float* a_raw   = ws + off; off += (size_t)ROWS_ * KG_;     // 10,485,760
  float* assign  = ws + off; off += (size_t)ROWS_ * K_;      //  8,388,608
  float* p_sum   = ws + off; off += (size_t)nblocks1 * KG_;  //     81,920
  float* p_ssq   = ws + off; off += (size_t)nblocks1 * KG_;  //     81,920
  float* ss      = ws + off; off += 2 * KG_;                 //        160
  float* a_sum   = ws + off; off += B_ * K_;                 //      4,096

  k1_gemm1<<<nblocks1, 256, 0, stream>>>(x, clusters, a_raw, p_sum, p_ssq);
  k2_bnstats<<<1, 96, 0, stream>>>(p_sum, p_ssq, bnw, bnb, ss, nblocks1);
  k3_softmax<<<ROWS_ / 256, 256, 0, stream>>>(a_raw, ss, assign);
  k3b_asum<<<B_, 256, 0, stream>>>(assign, a_sum);
  k4_vlad<<<B_ * 16, 256, 0, stream>>>(x, assign, a_sum, clusters2, out);
  k5_norm<<<B_, 256, 0, stream>>>(out);
}